// SAGELayer_55113020342353
// MI455X (gfx1250) — compile-verified
//
#include <hip/hip_runtime.h>
#include <hip/hip_bf16.h>

#define N_NODES 100000
#define N_EDGES 1600000
#define D 128
#define N_TILES (N_NODES / 16)   // 6250

typedef __attribute__((ext_vector_type(2))) float v2f;
typedef __attribute__((ext_vector_type(8))) float v8f;

// ---------------- zero workspace ----------------
__global__ __launch_bounds__(256) void sage_zero_kernel(float* __restrict__ p, int n) {
    int i = blockIdx.x * blockDim.x + threadIdx.x;
    if (i < n) p[i] = 0.0f;
}

// ---------------- edge scatter (mean-aggregate numerator + degree) ----------------
// 32 lanes per edge; each lane moves 4 floats of the 128-wide feature row.
__global__ __launch_bounds__(256) void sage_scatter_kernel(
    const float* __restrict__ x,
    const long long* __restrict__ edge_index,   // [2, N_EDGES] int64
    float* __restrict__ msg_sum,                // [N_NODES, D]
    float* __restrict__ deg)                    // [N_NODES]
{
    long long tid = (long long)blockIdx.x * blockDim.x + threadIdx.x;
    int e = (int)(tid >> 5);
    if (e >= N_EDGES) return;
    int lane = threadIdx.x & 31;

    int s = (int)edge_index[e];
    int d = (int)edge_index[e + N_EDGES];

    const float4 v = *reinterpret_cast<const float4*>(x + (long long)s * D + lane * 4);
    float* mp = msg_sum + (long long)d * D + lane * 4;
    atomicAdd(mp + 0, v.x);
    atomicAdd(mp + 1, v.y);
    atomicAdd(mp + 2, v.z);
    atomicAdd(mp + 3, v.w);
    if (lane == 0) atomicAdd(deg + d, 1.0f);
}

// ---------------- fused dual-GEMM + bias + L2 normalize + ReLU ----------------
// One wave per 16-node tile. out = (msg_sum/deg) @ Wl^T + x @ Wr^T + bl,
// then row-wise L2 normalize and ReLU, all in registers.
__global__ __launch_bounds__(256) void sage_gemm_kernel(
    const float* __restrict__ x,
    const float* __restrict__ msg_sum,
    const float* __restrict__ deg,
    const float* __restrict__ Wl,   // [D_OUT=128, D_IN=128]
    const float* __restrict__ bl,   // [128]
    const float* __restrict__ Wr,   // [128, 128]
    float* __restrict__ out)        // [N_NODES, 128]
{
    const int lane = threadIdx.x & 31;
    const int wave = (blockIdx.x * blockDim.x + threadIdx.x) >> 5;
    if (wave >= N_TILES) return;                 // wave-uniform: EXEC stays all-ones

    const int m  = lane & 15;                    // A row / B column within tile
    const int hi = lane >> 4;                    // half selector (K offset +2)
    const int node = wave * 16 + m;

    const float rdeg = 1.0f / fmaxf(deg[node], 1.0f);

    const float* __restrict__ xr = x       + node * D;
    const float* __restrict__ mr = msg_sum + node * D;
    // per-lane W row bases (column o = b*16 + m -> base + b*16*D)
    const float* __restrict__ wlr = Wl + m * D;
    const float* __restrict__ wrr = Wr + m * D;

    v8f acc[8];
#pragma unroll
    for (int b = 0; b < 8; ++b) acc[b] = {};

    for (int k0 = 0; k0 < D; k0 += 4) {
        const int kk = k0 + 2 * hi;

        // ---- batch-issue ALL fragment loads for this k-step into distinct regs ----
        v2f aL = *reinterpret_cast<const v2f*>(mr + kk);
        v2f aR = *reinterpret_cast<const v2f*>(xr + kk);
        v2f bLf[8], bRf[8];
#pragma unroll
        for (int b = 0; b < 8; ++b) {
            bLf[b] = *reinterpret_cast<const v2f*>(wlr + b * 16 * D + kk);
            bRf[b] = *reinterpret_cast<const v2f*>(wrr + b * 16 * D + kk);
        }
        aL[0] *= rdeg;
        aL[1] *= rdeg;

        // ---- drain in load order: scheduler can overlap loads with WMMAs ----
#pragma unroll
        for (int b = 0; b < 8; ++b) {
            acc[b] = __builtin_amdgcn_wmma_f32_16x16x4_f32(
                false, aL, false, bLf[b], (short)0, acc[b], false, false);
            acc[b] = __builtin_amdgcn_wmma_f32_16x16x4_f32(
                false, aR, false, bRf[b], (short)0, acc[b], false, false);
        }
    }

    // bias (depends only on column -> one scalar per block) + per-row sum of squares
    float ss[8];
#pragma unroll
    for (int r = 0; r < 8; ++r) ss[r] = 0.0f;
#pragma unroll
    for (int b = 0; b < 8; ++b) {
        const float bb = bl[b * 16 + m];
#pragma unroll
        for (int r = 0; r < 8; ++r) {
            float v = acc[b][r] + bb;
            acc[b][r] = v;
            ss[r] += v * v;
        }
    }

    // reduce across the 16 lanes of each half (C/D rows: M = r + 8*hi)
#pragma unroll
    for (int r = 0; r < 8; ++r) {
        float s = ss[r];
        s += __shfl_xor(s, 1, 32);
        s += __shfl_xor(s, 2, 32);
        s += __shfl_xor(s, 4, 32);
        s += __shfl_xor(s, 8, 32);
        const float scale = 1.0f / fmaxf(sqrtf(s), 1e-12f);
        const int row_node = wave * 16 + r + 8 * hi;
        float* __restrict__ orow = out + row_node * D;
#pragma unroll
        for (int b = 0; b < 8; ++b) {
            float v = acc[b][r] * scale;
            orow[b * 16 + m] = fmaxf(v, 0.0f);   // ReLU
        }
    }
}

extern "C" void kernel_launch(void* const* d_in, const int* in_sizes, int n_in,
                              void* d_out, int out_size, void* d_ws, size_t ws_size,
                              hipStream_t stream) {
    const float*     x  = (const float*)d_in[0];
    const long long* ei = (const long long*)d_in[1];
    // d_in[2] = batch (int64) -- unused by the reference computation
    const float*     Wl = (const float*)d_in[3];
    const float*     bl = (const float*)d_in[4];
    const float*     Wr = (const float*)d_in[5];
    float* out = (float*)d_out;

    float* msg_sum = (float*)d_ws;                  // N_NODES * D floats
    float* deg     = msg_sum + (size_t)N_NODES * D; // N_NODES floats

    // 1) zero msg_sum + deg
    {
        int n = N_NODES * D + N_NODES;
        int blocks = (n + 255) / 256;
        sage_zero_kernel<<<blocks, 256, 0, stream>>>(msg_sum, n);
    }
    // 2) scatter-add neighbor features + degrees
    {
        long long threads = (long long)N_EDGES * 32;
        int blocks = (int)((threads + 255) / 256);
        sage_scatter_kernel<<<blocks, 256, 0, stream>>>(x, ei, msg_sum, deg);
    }
    // 3) fused dual-GEMM (WMMA f32) + bias + L2-normalize + ReLU
    {
        int waves_per_block = 256 / 32;
        int blocks = (N_TILES + waves_per_block - 1) / waves_per_block;
        sage_gemm_kernel<<<blocks, 256, 0, stream>>>(x, msg_sum, deg, Wl, bl, Wr, out);
    }
}